// BasicCL4CTR_49727131353769
// MI455X (gfx1250) — compile-verified
//
#include <hip/hip_runtime.h>
#include <hip/hip_bf16.h>

#define B_TOT   4096
#define NF      39
#define FPAD    48
#define ED      16
#define EPS_U   1e-4f
#define BETA_S  0.01f
#define WPB     8   // waves (samples) per 256-thread block

typedef __attribute__((ext_vector_type(2))) float v2f;
typedef __attribute__((ext_vector_type(8))) float v8f;

__global__ void cl4ctr_zero_ws(float* __restrict__ ws, int n) {
    int i = blockIdx.x * blockDim.x + threadIdx.x;
    if (i < n) ws[i] = 0.f;
}

__global__ __launch_bounds__(256) void cl4ctr_main(
    const int* __restrict__ x, const float* __restrict__ emb,
    float* __restrict__ s_acc, float* __restrict__ scal_acc)
{
    __shared__ float sE[WPB][FPAD * ED];   // per-wave padded 48x16 tile
    __shared__ float sNorm[WPB][FPAD];     // per-wave row norms
    __shared__ float sS[NF * ED];          // block accumulator for batch-sum s
    __shared__ float sScal[2];             // block accumulators: sumsq, uniform

    const int lane = threadIdx.x & 31;
    const int wave = threadIdx.x >> 5;
    const int b    = blockIdx.x * WPB + wave;

    for (int i = threadIdx.x; i < NF * ED; i += 256) sS[i] = 0.f;
    if (threadIdx.x < 2) sScal[threadIdx.x] = 0.f;

    float* E = sE[wave];

    // ---- gather: 48 rows, 8 rows/iter, 4 lanes per row each moving a float4
    const int rsub = lane >> 2;           // 0..7
    const int c4   = (lane & 3) << 2;     // 0,4,8,12
    for (int it = 0; it < 6; ++it) {
        int f = it * 8 + rsub;            // 0..47
        float4 v = make_float4(0.f, 0.f, 0.f, 0.f);
        if (f < NF) {
            int gidx = x[b * NF + f] + f * 100000;   // per-field offset (all dims 100000)
            v = *(const float4*)(emb + (size_t)gidx * ED + c4);
        }
        *(float4*)(&E[f * ED + c4]) = v;
    }
    __syncthreads();

    // ---- norms, sum of squared norms, block-level s accumulation
    float sumsq = 0.f;
    for (int r = lane; r < FPAD; r += 32) {
        float acc = 0.f;
        #pragma unroll
        for (int d = 0; d < ED; ++d) { float e = E[r * ED + d]; acc += e * e; }
        sNorm[wave][r] = __builtin_amdgcn_sqrtf(acc);      // v_sqrt_f32
        if (r < NF) {
            sumsq += acc;
            #pragma unroll
            for (int d = 0; d < ED; ++d) atomicAdd(&sS[r * ED + d], E[r * ED + d]);
        }
    }
    __syncthreads();

    // ---- WMMA fragments: 16x4 f32 A-layout; for a Gram, A and B frags coincide.
    // lanes 0-15: M=lane, {K=kc+0, kc+1}; lanes 16-31: M=lane-16, {K=kc+2, kc+3}
    v2f fr[3][4];
    const int m     = lane & 15;
    const int khalf = (lane >> 4) << 1;
    #pragma unroll
    for (int t = 0; t < 3; ++t)
        #pragma unroll
        for (int kc = 0; kc < 4; ++kc) {
            const float* p = &E[(t * 16 + m) * ED + kc * 4 + khalf];
            fr[t][kc].x = p[0];
            fr[t][kc].y = p[1];
        }

    // ---- 6 symmetric tiles of the 48x48 (padded) Gram, K=16 via 4 x wmma f32 16x16x4
    float uni = 0.f;
    #pragma unroll
    for (int ti = 0; ti < 3; ++ti) {
        #pragma unroll
        for (int tj = ti; tj < 3; ++tj) {
            v8f c = {};
            #pragma unroll
            for (int kc = 0; kc < 4; ++kc)
                c = __builtin_amdgcn_wmma_f32_16x16x4_f32(
                        false, fr[ti][kc], false, fr[tj][kc],
                        (short)0, c, false, false);
            const float w  = (ti == tj) ? 1.f : 2.f;
            const float nj = sNorm[wave][tj * 16 + m];          // N = lane&15
            const int   mb = ti * 16 + ((lane >> 4) << 3);      // M = r (+8 for hi half)
            #pragma unroll
            for (int r = 0; r < 8; ++r) {
                float ni = sNorm[wave][mb + r];
                // hardware reciprocal: one v_rcp_f32 + fma instead of IEEE div sequence
                float rd = __builtin_amdgcn_rcpf(ni * nj + EPS_U);
                uni = __builtin_fmaf(w * c[r], rd, uni);        // padded rows: 0*rcp(EPS)=0
            }
        }
    }

    // ---- wave32 reduction, then block accumulation
    for (int o = 16; o > 0; o >>= 1) {
        uni   += __shfl_xor(uni,   o, 32);
        sumsq += __shfl_xor(sumsq, o, 32);
    }
    if (lane == 0) {
        atomicAdd(&sScal[0], sumsq);
        atomicAdd(&sScal[1], uni);
    }
    __syncthreads();

    // ---- one set of global atomics per block
    if (threadIdx.x < 2) atomicAdd(&scal_acc[threadIdx.x], sScal[threadIdx.x]);
    for (int i = threadIdx.x; i < NF * ED; i += 256) atomicAdd(&s_acc[i], sS[i]);
}

__global__ void cl4ctr_finalize(const float* __restrict__ s_acc,
                                const float* __restrict__ scal_acc,
                                float* __restrict__ out)
{
    __shared__ float red[256];
    float a = 0.f;
    for (int i = threadIdx.x; i < NF * ED; i += 256) { float v = s_acc[i]; a += v * v; }
    red[threadIdx.x] = a;
    __syncthreads();
    for (int s = 128; s > 0; s >>= 1) {
        if (threadIdx.x < s) red[threadIdx.x] += red[threadIdx.x + s];
        __syncthreads();
    }
    if (threadIdx.x == 0) {
        float sumsq    = scal_acc[0];
        float uni      = scal_acc[1];
        float n_pairs  = 0.5f * (float)B_TOT * (float)(B_TOT - 1);
        float pair_sum = (float)B_TOT * sumsq - red[0];
        float align    = pair_sum / (n_pairs * (float)NF);
        float uniform  = uni / ((float)B_TOT * (float)NF * (float)NF);
        out[0] = (align + uniform) * BETA_S;
    }
}

extern "C" void kernel_launch(void* const* d_in, const int* in_sizes, int n_in,
                              void* d_out, int out_size, void* d_ws, size_t ws_size,
                              hipStream_t stream) {
    const int*   x   = (const int*)d_in[0];     // [4096, 39] int32
    const float* emb = (const float*)d_in[1];   // [3,900,000, 16] f32
    float* ws       = (float*)d_ws;
    float* s_acc    = ws;                       // 39*16 floats
    float* scal_acc = ws + NF * ED;             // 2 floats: {sumsq, uniform}

    cl4ctr_zero_ws<<<3, 256, 0, stream>>>(ws, NF * ED + 2);
    cl4ctr_main<<<B_TOT / WPB, 256, 0, stream>>>(x, emb, s_acc, scal_acc);
    cl4ctr_finalize<<<1, 256, 0, stream>>>(s_acc, scal_acc, (float*)d_out);
}